// Net_45286135169541
// MI455X (gfx1250) — compile-verified
//
#include <hip/hip_runtime.h>
#include <hip/hip_bf16.h>

// ---------------- problem constants ----------------
#define Bn     64
#define IN_C   4
#define Vn     4096
#define V2n    1024
#define V3n    256
#define F1n    32          // CL1_F
#define F2n    64          // CL2_F
#define Kch    25          // Chebyshev order
#define E0n    65536       // Vn*16
#define E1n    16384       // V2n*16
#define CB0    (IN_C*Bn)   // 256  cols of level-0 state
#define CB1    (F1n*Bn)    // 2048 cols of level-1 state
#define FC1IN  (F2n*V3n)   // 16384
#define FC1F   512
#define FC2F   10
#define FC1_SPLIT 8        // split-K factor for FC1
#define GEMM_MT   4        // M-tiles per wave in the Chebyshev GEMM

typedef __attribute__((ext_vector_type(2))) float v2f;
typedef __attribute__((ext_vector_type(8))) float v8f;

// true fp32 matrix op: D = A(16x4) * B(4x16) + C  -> v_wmma_f32_16x16x4_f32
__device__ __forceinline__ v8f wmma_f32_k4(v2f a, v2f b, v8f c) {
    return __builtin_amdgcn_wmma_f32_16x16x4_f32(
        /*neg_a=*/false, a, /*neg_b=*/false, b,
        /*c_mod=*/(short)0, c, /*reuse_a=*/false, /*reuse_b=*/false);
}

// ---------------- elementwise / data movement ----------------

// x (B, C, V) -> x0 (V, C*B) : x0[v, c*B+b] = x[b,c,v]
__global__ void k_transpose_x0(const float* __restrict__ x, float* __restrict__ x0) {
    int tid = blockIdx.x * blockDim.x + threadIdx.x;
    if (tid >= Vn * CB0) return;
    int v = tid / CB0;
    int r = tid - v * CB0;
    int c = r / Bn, b = r - c * Bn;
    x0[tid] = x[((size_t)b * IN_C + c) * Vn + v];
}

// dst = coef * src, float4-vectorized (n multiple of 4)
__global__ void k_scale_copy4(const float* __restrict__ src, float* __restrict__ dst,
                              float coef, int n4) {
    int i = blockIdx.x * blockDim.x + threadIdx.x;
    if (i >= n4) return;
    float4 s = ((const float4*)src)[i];
    s.x *= coef; s.y *= coef; s.z *= coef; s.w *= coef;
    ((float4*)dst)[i] = s;
}

// COO scatter, 4 columns per thread:
//   dst[rows[e], j..j+3] += scale*vals[e] * src[cols[e], j..j+3]
__global__ void k_spmm_scatter4(const int* __restrict__ rows, const int* __restrict__ cols,
                                const float* __restrict__ vals,
                                const float* __restrict__ src, float* __restrict__ dst,
                                float scale, int E, int ncols) {
    int nq = ncols >> 2;
    int tid = blockIdx.x * blockDim.x + threadIdx.x;
    if (tid >= E * nq) return;
    int e  = tid / nq;
    int j  = (tid - e * nq) << 2;
    float a = scale * vals[e];
    float4 s = *(const float4*)(src + (size_t)cols[e] * ncols + j);  // b128 load
    float* d = dst + (size_t)rows[e] * ncols + j;
    atomicAdd(d + 0, a * s.x);
    atomicAdd(d + 1, a * s.y);
    atomicAdd(d + 2, a * s.z);
    atomicAdd(d + 3, a * s.w);
}

// relu+maxpool4 over v, re-layout for next conv; 4 f-values per thread:
// x1[v2, f*B + b] = relu(max_q Y1[(b*V + 4*v2+q), f])
__global__ void k_relu_pool_tr1(const float* __restrict__ Y1, float* __restrict__ x1) {
    int tid = blockIdx.x * blockDim.x + threadIdx.x;        // over V2n*Bn*(F1n/4)
    const int FQ = F1n / 4;
    if (tid >= V2n * Bn * FQ) return;
    int v2 = tid / (Bn * FQ);
    int r  = tid - v2 * (Bn * FQ);
    int b  = r / FQ;
    int f4 = (r - b * FQ) << 2;
    const float* p = Y1 + ((size_t)b * Vn + 4 * v2) * F1n + f4;
    float4 m  = *(const float4*)(p);
    float4 m1 = *(const float4*)(p + F1n);
    float4 m2 = *(const float4*)(p + 2 * F1n);
    float4 m3 = *(const float4*)(p + 3 * F1n);
    m.x = fmaxf(fmaxf(fmaxf(m.x, m1.x), fmaxf(m2.x, m3.x)), 0.0f);
    m.y = fmaxf(fmaxf(fmaxf(m.y, m1.y), fmaxf(m2.y, m3.y)), 0.0f);
    m.z = fmaxf(fmaxf(fmaxf(m.z, m1.z), fmaxf(m2.z, m3.z)), 0.0f);
    m.w = fmaxf(fmaxf(fmaxf(m.w, m1.w), fmaxf(m2.w, m3.w)), 0.0f);
    float* o = x1 + (size_t)v2 * CB1 + (size_t)f4 * Bn + b;
    o[0]      = m.x;
    o[Bn]     = m.y;
    o[2 * Bn] = m.z;
    o[3 * Bn] = m.w;
}

// relu+maxpool4 + flatten; 4 f-values per thread:
// feat[b, f*256 + v3] = relu(max_q Y2[(b*V2 + 4*v3+q), f])
__global__ void k_relu_pool_flat2(const float* __restrict__ Y2, float* __restrict__ feat) {
    int tid = blockIdx.x * blockDim.x + threadIdx.x;        // over Bn*V3n*(F2n/4)
    const int FQ = F2n / 4;
    if (tid >= Bn * V3n * FQ) return;
    int b  = tid / (V3n * FQ);
    int r  = tid - b * (V3n * FQ);
    int v3 = r / FQ;
    int f4 = (r - v3 * FQ) << 2;
    const float* p = Y2 + ((size_t)b * V2n + 4 * v3) * F2n + f4;
    float4 m  = *(const float4*)(p);
    float4 m1 = *(const float4*)(p + F2n);
    float4 m2 = *(const float4*)(p + 2 * F2n);
    float4 m3 = *(const float4*)(p + 3 * F2n);
    m.x = fmaxf(fmaxf(fmaxf(m.x, m1.x), fmaxf(m2.x, m3.x)), 0.0f);
    m.y = fmaxf(fmaxf(fmaxf(m.y, m1.y), fmaxf(m2.y, m3.y)), 0.0f);
    m.z = fmaxf(fmaxf(fmaxf(m.z, m1.z), fmaxf(m2.z, m3.z)), 0.0f);
    m.w = fmaxf(fmaxf(fmaxf(m.w, m1.w), fmaxf(m2.w, m3.w)), 0.0f);
    float* o = feat + (size_t)b * FC1IN + (size_t)f4 * V3n + v3;
    o[0]       = m.x;
    o[V3n]     = m.y;
    o[2 * V3n] = m.z;
    o[3 * V3n] = m.w;
}

// ---------------- WMMA kernels ----------------

// Per-Chebyshev-step dense accumulation:
//   Y[b*Vdim+v, f] += sum_c xs_k[v, c*B+b] * W[f, c*Kch + k]
// One wave handles GEMM_MT vertical 16x16 tiles (64 rows x 16 f), sharing the
// B operand across the 4 back-to-back v_wmma_f32_16x16x4_f32 per K-chunk.
__global__ void k_gemm_cheb_acc(const float* __restrict__ xs, const float* __restrict__ W,
                                const float* __restrict__ bias, float* __restrict__ Y,
                                int Vdim, int C, int Fout, int k, int init) {
    int wave = (blockIdx.x * blockDim.x + threadIdx.x) >> 5;
    int lane = threadIdx.x & 31;
    int tiles_n = Fout >> 4;
    int rows    = Bn * Vdim;
    int mgroups = rows / (16 * GEMM_MT);
    if (wave >= mgroups * tiles_n) return;
    int tm = wave / tiles_n, tn = wave - tm * tiles_n;
    int nbase = tm * (16 * GEMM_MT), fbase = tn << 4;

    int half  = lane >> 4;       // 0: K pair {0,1}, 1: K pair {2,3}
    int l15   = lane & 15;
    int kksel = half << 1;

    // A operand row pointers (per M-tile): xs[v, c*B+b], +c*Bn per channel
    const float* xsrow[GEMM_MT];
    #pragma unroll
    for (int t = 0; t < GEMM_MT; ++t) {
        int nA = nbase + t * 16 + l15;
        int bA = nA / Vdim, vA = nA - bA * Vdim;
        xsrow[t] = xs + (size_t)vA * ((size_t)C * Bn) + bA;
    }
    // B operand: W[f, c*Kch + k], +c*Kch per channel
    int f = fbase + l15;
    const float* Wrow = W + (size_t)f * ((size_t)C * Kch) + k;

    v8f acc[GEMM_MT];
    if (init) {
        float bv = bias[f];
        #pragma unroll
        for (int t = 0; t < GEMM_MT; ++t)
            #pragma unroll
            for (int r = 0; r < 8; ++r) acc[t][r] = bv;
    } else {
        #pragma unroll
        for (int t = 0; t < GEMM_MT; ++t)
            #pragma unroll
            for (int r = 0; r < 8; ++r) {
                int M = nbase + t * 16 + r + (half << 3);
                acc[t][r] = Y[(size_t)M * Fout + f];
            }
    }

    for (int c0 = 0; c0 < C; c0 += 4) {
        v2f bb;
        bb.x = Wrow[(size_t)(c0 + kksel)     * Kch];
        bb.y = Wrow[(size_t)(c0 + kksel + 1) * Kch];
        #pragma unroll
        for (int t = 0; t < GEMM_MT; ++t) {
            v2f a;
            a.x = xsrow[t][(size_t)(c0 + kksel)     * Bn];
            a.y = xsrow[t][(size_t)(c0 + kksel + 1) * Bn];
            acc[t] = wmma_f32_k4(a, bb, acc[t]);
        }
    }

    #pragma unroll
    for (int t = 0; t < GEMM_MT; ++t)
        #pragma unroll
        for (int r = 0; r < 8; ++r) {
            int M = nbase + t * 16 + r + (half << 3);
            Y[(size_t)M * Fout + f] = acc[t][r];
        }
}

// FC1 accumulator init: fc1o[b, j] = bf1[j]
__global__ void k_fc1_init(const float* __restrict__ bf1, float* __restrict__ fc1o) {
    int t = blockIdx.x * blockDim.x + threadIdx.x;
    if (t >= Bn * FC1F) return;
    fc1o[t] = bf1[t & (FC1F - 1)];
}

// FC1 split-K: each wave owns one 16x16 tile and one 2048-deep K chunk;
// fp32 WMMA down the chunk, partials combined via global_atomic_add_f32.
__global__ void k_fc1_wmma_splitk(const float* __restrict__ feat,
                                  const float* __restrict__ Wf1,
                                  float* __restrict__ fc1o) {
    int wave = (blockIdx.x * blockDim.x + threadIdx.x) >> 5;
    int lane = threadIdx.x & 31;
    const int tiles_n = FC1F / 16;                            // 32
    const int tiles   = (Bn / 16) * tiles_n;                  // 128
    if (wave >= tiles * FC1_SPLIT) return;
    int tile = wave / FC1_SPLIT;
    int sk   = wave - tile * FC1_SPLIT;
    int tm = tile / tiles_n, tn = tile - tm * tiles_n;
    int bbase = tm << 4, jbase = tn << 4;
    const int KC = FC1IN / FC1_SPLIT;                         // 2048
    int i0 = sk * KC;

    int half = lane >> 4, l15 = lane & 15, kksel = half << 1;
    const float* arow = feat + (size_t)(bbase + l15) * FC1IN + i0;
    const float* brow = Wf1  + (size_t)(jbase + l15) * FC1IN + i0;

    v8f acc;
    #pragma unroll
    for (int r = 0; r < 8; ++r) acc[r] = 0.0f;

    for (int i = 0; i < KC; i += 4) {
        __builtin_prefetch(arow + i + 256, 0, 1);
        __builtin_prefetch(brow + i + 256, 0, 1);
        float2 av = *(const float2*)(arow + i + kksel);       // contiguous pair -> b64 load
        float2 wv = *(const float2*)(brow + i + kksel);
        v2f a, b;
        a.x = av.x; a.y = av.y;
        b.x = wv.x; b.y = wv.y;
        acc = wmma_f32_k4(a, b, acc);
    }

    #pragma unroll
    for (int r = 0; r < 8; ++r) {
        int M = r + (half << 3);
        atomicAdd(&fc1o[(size_t)(bbase + M) * FC1F + jbase + l15], acc[r]);
    }
}

// in-place ReLU on the FC1 accumulator
__global__ void k_relu_inplace(float* __restrict__ p, int n) {
    int i = blockIdx.x * blockDim.x + threadIdx.x;
    if (i < n) p[i] = fmaxf(p[i], 0.0f);
}

// FC2: tiny (64x512 @ 512x10) — plain VALU with float4 loads
__global__ void k_fc2(const float* __restrict__ h, const float* __restrict__ Wf2,
                      const float* __restrict__ bf2, float* __restrict__ out) {
    int t = blockIdx.x * blockDim.x + threadIdx.x;
    if (t >= Bn * FC2F) return;
    int b = t / FC2F, j = t - b * FC2F;
    const float4* hr = (const float4*)(h + (size_t)b * FC1F);
    const float4* wr = (const float4*)(Wf2 + (size_t)j * FC1F);
    float s = bf2[j];
    for (int i = 0; i < FC1F / 4; ++i) {
        float4 hv = hr[i], wv = wr[i];
        s += hv.x * wv.x + hv.y * wv.y + hv.z * wv.z + hv.w * wv.w;
    }
    out[t] = s;
}

// ---------------- host orchestration ----------------
static inline int blk(long long n, int bs) { return (int)((n + bs - 1) / bs); }

extern "C" void kernel_launch(void* const* d_in, const int* in_sizes, int n_in,
                              void* d_out, int out_size, void* d_ws, size_t ws_size,
                              hipStream_t stream) {
    const float* x     = (const float*)d_in[0];
    const int*   rows0 = (const int*)  d_in[1];
    const int*   cols0 = (const int*)  d_in[2];
    const float* vals0 = (const float*)d_in[3];
    const int*   rows1 = (const int*)  d_in[4];
    const int*   cols1 = (const int*)  d_in[5];
    const float* vals1 = (const float*)d_in[6];
    const float* W1  = (const float*)d_in[7];
    const float* b1  = (const float*)d_in[8];
    const float* W2  = (const float*)d_in[9];
    const float* b2  = (const float*)d_in[10];
    const float* Wf1 = (const float*)d_in[11];
    const float* bf1 = (const float*)d_in[12];
    const float* Wf2 = (const float*)d_in[13];
    const float* bf2 = (const float*)d_in[14];
    float* out = (float*)d_out;

    // workspace carve-up (~92.4 MB of fp32)
    float* ws = (float*)d_ws;
    size_t off = 0;
    const size_t S0 = (size_t)Vn * CB0;    // 1,048,576
    const size_t S1 = (size_t)V2n * CB1;   // 2,097,152
    float* buf0[3]; for (int i = 0; i < 3; ++i) { buf0[i] = ws + off; off += S0; }
    float* Y1 = ws + off; off += (size_t)Bn * Vn * F1n;       // 8,388,608
    float* buf1[3]; for (int i = 0; i < 3; ++i) { buf1[i] = ws + off; off += S1; }
    float* Y2   = ws + off; off += (size_t)Bn * V2n * F2n;    // 4,194,304
    float* feat = ws + off; off += (size_t)Bn * FC1IN;        // 1,048,576
    float* fc1o = ws + off; off += (size_t)Bn * FC1F;         //    32,768
    (void)ws_size; (void)in_sizes; (void)n_in; (void)out_size;

    const int BS = 256;

    // ---- level 0: graph conv (C=4, F=32, K=25) ----
    k_transpose_x0<<<blk((long long)S0, BS), BS, 0, stream>>>(x, buf0[0]);

    long long waves0 = ((long long)Bn * Vn / (16 * GEMM_MT)) * (F1n / 16);  // 8192
    int g0 = blk(waves0 * 32, BS);
    k_gemm_cheb_acc<<<g0, BS, 0, stream>>>(buf0[0], W1, b1, Y1, Vn, IN_C, F1n, 0, 1);

    {
        float* prev = buf0[0];   // T_0
        float* cur  = buf0[1];
        float* nxt  = buf0[2];
        // k = 1 : T_1 = L @ T_0
        k_scale_copy4<<<blk((long long)S0 / 4, BS), BS, 0, stream>>>(prev, cur, 0.0f, (int)(S0 / 4));
        k_spmm_scatter4<<<blk((long long)E0n * (CB0 / 4), BS), BS, 0, stream>>>(
            rows0, cols0, vals0, prev, cur, 1.0f, E0n, CB0);
        k_gemm_cheb_acc<<<g0, BS, 0, stream>>>(cur, W1, b1, Y1, Vn, IN_C, F1n, 1, 0);
        // k >= 2 : T_k = 2 L T_{k-1} - T_{k-2}
        for (int k = 2; k < Kch; ++k) {
            k_scale_copy4<<<blk((long long)S0 / 4, BS), BS, 0, stream>>>(prev, nxt, -1.0f, (int)(S0 / 4));
            k_spmm_scatter4<<<blk((long long)E0n * (CB0 / 4), BS), BS, 0, stream>>>(
                rows0, cols0, vals0, cur, nxt, 2.0f, E0n, CB0);
            k_gemm_cheb_acc<<<g0, BS, 0, stream>>>(nxt, W1, b1, Y1, Vn, IN_C, F1n, k, 0);
            float* t = prev; prev = cur; cur = nxt; nxt = t;
        }
    }

    // relu + maxpool4 + transpose into level-1 state
    k_relu_pool_tr1<<<blk((long long)V2n * Bn * (F1n / 4), BS), BS, 0, stream>>>(Y1, buf1[0]);

    // ---- level 1: graph conv (C=32, F=64, K=25) ----
    long long waves1 = ((long long)Bn * V2n / (16 * GEMM_MT)) * (F2n / 16);  // 4096
    int g1 = blk(waves1 * 32, BS);
    k_gemm_cheb_acc<<<g1, BS, 0, stream>>>(buf1[0], W2, b2, Y2, V2n, F1n, F2n, 0, 1);

    {
        float* prev = buf1[0];
        float* cur  = buf1[1];
        float* nxt  = buf1[2];
        k_scale_copy4<<<blk((long long)S1 / 4, BS), BS, 0, stream>>>(prev, cur, 0.0f, (int)(S1 / 4));
        k_spmm_scatter4<<<blk((long long)E1n * (CB1 / 4), BS), BS, 0, stream>>>(
            rows1, cols1, vals1, prev, cur, 1.0f, E1n, CB1);
        k_gemm_cheb_acc<<<g1, BS, 0, stream>>>(cur, W2, b2, Y2, V2n, F1n, F2n, 1, 0);
        for (int k = 2; k < Kch; ++k) {
            k_scale_copy4<<<blk((long long)S1 / 4, BS), BS, 0, stream>>>(prev, nxt, -1.0f, (int)(S1 / 4));
            k_spmm_scatter4<<<blk((long long)E1n * (CB1 / 4), BS), BS, 0, stream>>>(
                rows1, cols1, vals1, cur, nxt, 2.0f, E1n, CB1);
            k_gemm_cheb_acc<<<g1, BS, 0, stream>>>(nxt, W2, b2, Y2, V2n, F1n, F2n, k, 0);
            float* t = prev; prev = cur; cur = nxt; nxt = t;
        }
    }

    // relu + maxpool4 + flatten
    k_relu_pool_flat2<<<blk((long long)Bn * V3n * (F2n / 4), BS), BS, 0, stream>>>(Y2, feat);

    // FC1: bias-init, split-K fp32 WMMA with atomic combine, then ReLU
    k_fc1_init<<<blk((long long)Bn * FC1F, BS), BS, 0, stream>>>(bf1, fc1o);
    k_fc1_wmma_splitk<<<blk(128LL * FC1_SPLIT * 32, BS), BS, 0, stream>>>(feat, Wf1, fc1o);
    k_relu_inplace<<<blk((long long)Bn * FC1F, BS), BS, 0, stream>>>(fc1o, Bn * FC1F);

    // FC2
    k_fc2<<<blk((long long)Bn * FC2F, BS), BS, 0, stream>>>(fc1o, Wf2, bf2, out);
}